// Conv2dFTN_76020921140169
// MI455X (gfx1250) — compile-verified
//
#include <hip/hip_runtime.h>
#include <math.h>

#define BINS   2112
#define NSZ    64
#define HALF   33
#define BATCH  64
#define PI_F   3.14159265358979f

typedef __attribute__((ext_vector_type(2))) float v2f;
typedef __attribute__((ext_vector_type(8))) float v8f;

// ---------------------------------------------------------------------------
// 1) Weight DFT: Ff[bin,o,i] = exp(-2pi*i*(u+v)/64) * conj(DFT9(F[o,i]))
//    + Frobenius-norm accumulation (block reduce + one atomic per block)
// ---------------------------------------------------------------------------
__global__ __launch_bounds__(256)
void weight_dft_kernel(const float* __restrict__ F, int co, int ci,
                       float* __restrict__ FfR, float* __restrict__ FfI,
                       float* __restrict__ normAcc)
{
    __shared__ float red[256];
    int total = BINS * co * ci;
    int idx = blockIdx.x * 256 + threadIdx.x;
    float s2 = 0.f;
    if (idx < total) {
        int bin = idx / (co * ci);
        int rem = idx % (co * ci);
        int u = bin / HALF, v = bin % HALF;
        float fr = 0.f, fi = 0.f;
        #pragma unroll
        for (int a = 0; a < 3; a++)
            #pragma unroll
            for (int b = 0; b < 3; b++) {
                float w  = F[rem * 9 + a * 3 + b];
                float th = -2.f * PI_F * (float)(u * a + v * b) / 64.f;
                fr += w * cosf(th);
                fi += w * sinf(th);
            }
        fi = -fi;  // conj
        float th = -2.f * PI_F * (float)(u + v) / 64.f;  // shift sm, s=1
        float cr = cosf(th), cs = sinf(th);
        float rr = fr * cr - fi * cs;
        float ri = fr * cs + fi * cr;
        FfR[idx] = rr; FfI[idx] = ri;
        s2 = rr * rr + ri * ri;
    }
    red[threadIdx.x] = s2;
    __syncthreads();
    for (int s = 128; s > 0; s >>= 1) {
        if (threadIdx.x < s) red[threadIdx.x] += red[threadIdx.x + s];
        __syncthreads();
    }
    if (threadIdx.x == 0) atomicAdd(normAcc, red[0]);
}

// ---------------------------------------------------------------------------
// 2) Cayley per bin: A = U - U^H + V^H V  (64x64 cplx), inv(I+A) via
//    Gauss-Jordan in LDS, out = [inv@(I-A) ; -2 V@inv].  ci==64 always.
//    transIn : input stored (64 x CO) instead of (CO x 64)
//    transOut: write result transposed (64 x CO layout)
// ---------------------------------------------------------------------------
#define CAYLEY_LDS_FLOATS (4096*2 + 8192*2 + 64*2 + 2)
#define CAYLEY_LDS_BYTES  (CAYLEY_LDS_FLOATS * 4)

__global__ __launch_bounds__(256)
void cayley_kernel(const float* __restrict__ FfR, const float* __restrict__ FfI,
                   const float* __restrict__ fptr, const float* __restrict__ normAcc,
                   int CO, int transIn, int transOut,
                   float* __restrict__ outR, float* __restrict__ outI)
{
    extern __shared__ float lds[];
    float* Ar   = lds;            // 4096
    float* Ai   = Ar + 4096;      // 4096
    float* Gr   = Ai + 4096;      // 8192 (64 x 128 augmented)
    float* Gi   = Gr + 8192;      // 8192
    float* colr = Gi + 8192;      // 64
    float* coli = colr + 64;      // 64
    float* piv  = coli + 64;      // 2

    int bin = blockIdx.x;
    int tid = threadIdx.x;
    float scale = fptr[0] / sqrtf(normAcc[0]);
    size_t base = (size_t)bin * CO * 64;

    auto loadW = [&](int row, int col, float& wr, float& wi) {
        size_t idx = transIn ? (base + (size_t)col * CO + row)
                             : (base + (size_t)row * 64 + col);
        wr = scale * FfR[idx];
        wi = scale * FfI[idx];
    };

    // A = U - U^H + V^H V
    for (int e = tid; e < 4096; e += 256) {
        int m = e >> 6, j = e & 63;
        float wmr, wmi, wjr, wji;
        loadW(m, j, wmr, wmi);
        loadW(j, m, wjr, wji);
        float arr = wmr - wjr;
        float aii = wmi + wji;
        for (int r = 64; r < CO; r++) {
            float vr1, vi1, vr2, vi2;
            loadW(r, m, vr1, vi1);
            loadW(r, j, vr2, vi2);
            arr += vr1 * vr2 + vi1 * vi2;   // conj(v1)*v2
            aii += vr1 * vi2 - vi1 * vr2;
        }
        Ar[e] = arr; Ai[e] = aii;
    }
    __syncthreads();

    // G = [ I+A | I ]
    for (int e = tid; e < 8192; e += 256) {
        int m = e >> 7, c = e & 127;
        float gr, gi;
        if (c < 64) { gr = Ar[m * 64 + c] + ((m == c) ? 1.f : 0.f); gi = Ai[m * 64 + c]; }
        else        { gr = ((c - 64) == m) ? 1.f : 0.f;             gi = 0.f; }
        Gr[e] = gr; Gi[e] = gi;
    }
    __syncthreads();

    // Gauss-Jordan
    for (int p = 0; p < 64; p++) {
        if (tid < 64) { colr[tid] = Gr[tid * 128 + p]; coli[tid] = Gi[tid * 128 + p]; }
        __syncthreads();
        if (tid == 0) {
            float dr = colr[p], di = coli[p];
            float d  = dr * dr + di * di;
            piv[0] = dr / d; piv[1] = -di / d;
        }
        __syncthreads();
        for (int c = tid; c < 128; c += 256) {
            float gr = Gr[p * 128 + c], gi = Gi[p * 128 + c];
            Gr[p * 128 + c] = gr * piv[0] - gi * piv[1];
            Gi[p * 128 + c] = gr * piv[1] + gi * piv[0];
        }
        __syncthreads();
        for (int e = tid; e < 8192; e += 256) {
            int m = e >> 7, c = e & 127;
            if (m == p) continue;
            float mr = colr[m], mi = coli[m];
            float pr = Gr[p * 128 + c], pi2 = Gi[p * 128 + c];
            Gr[e] -= mr * pr - mi * pi2;
            Gi[e] -= mr * pi2 + mi * pr;
        }
        __syncthreads();
    }

    auto storeOut = [&](int row, int col, float vr, float vi) {
        size_t idx = transOut ? (base + (size_t)col * CO + row)
                              : (base + (size_t)row * 64 + col);
        outR[idx] = vr; outI[idx] = vi;
    };

    // O1 = inv @ (I - A)
    for (int e = tid; e < 4096; e += 256) {
        int m = e >> 6, j = e & 63;
        float accr = Gr[m * 128 + 64 + j];
        float acci = Gi[m * 128 + 64 + j];
        for (int k = 0; k < 64; k++) {
            float ir = Gr[m * 128 + 64 + k], ii = Gi[m * 128 + 64 + k];
            float akr = Ar[k * 64 + j],      aki = Ai[k * 64 + j];
            accr -= ir * akr - ii * aki;
            acci -= ir * aki + ii * akr;
        }
        storeOut(m, j, accr, acci);
    }
    // O2 = -2 V @ inv
    int vrows = CO - 64;
    for (int e = tid; e < vrows * 64; e += 256) {
        int r = e >> 6, j = e & 63;
        float accr = 0.f, acci = 0.f;
        for (int k = 0; k < 64; k++) {
            float vr1, vi1;
            loadW(64 + r, k, vr1, vi1);
            float ir = Gr[k * 128 + 64 + j], ii = Gi[k * 128 + 64 + j];
            accr += vr1 * ir - vi1 * ii;
            acci += vr1 * ii + vi1 * ir;
        }
        storeOut(64 + r, j, -2.f * accr, -2.f * acci);
    }
}

// ---------------------------------------------------------------------------
// 3) rfft2 (real 64x64 -> 64x33 half-spectrum), one block per (b,ch) image.
//    Output layout (bins, CH, BATCH), split re/im.
// ---------------------------------------------------------------------------
__global__ __launch_bounds__(256)
void rfft2_fwd_kernel(const float* __restrict__ src, int CH, float scale,
                      float* __restrict__ outR, float* __restrict__ outI)
{
    __shared__ float img[4096];
    __shared__ float z1r[2112], z1i[2112];
    __shared__ float tc[64], ts[64];
    int b = blockIdx.x / CH, ch = blockIdx.x % CH;
    int tid = threadIdx.x;
    if (tid < 64) {
        float th = 2.f * PI_F * (float)tid / 64.f;
        tc[tid] = cosf(th); ts[tid] = sinf(th);
    }
    const float* sp = src + ((size_t)(b * CH + ch)) * 4096;
    for (int i = tid; i < 4096; i += 256) img[i] = scale * sp[i];
    __syncthreads();
    // rows: Z1[r][v] = sum_c img[r][c] * e^{-2pi i c v / 64}
    for (int i = tid; i < 2112; i += 256) {
        int r = i / HALF, v = i % HALF;
        float sr = 0.f, si = 0.f;
        for (int c = 0; c < 64; c++) {
            int t = (c * v) & 63;
            float x = img[r * 64 + c];
            sr += x * tc[t];
            si -= x * ts[t];
        }
        z1r[i] = sr; z1i[i] = si;
    }
    __syncthreads();
    // cols: S[u][v] = sum_r Z1[r][v] * e^{-2pi i r u / 64}
    for (int i = tid; i < 2112; i += 256) {
        int u = i / HALF, v = i % HALF;
        float sr = 0.f, si = 0.f;
        for (int r = 0; r < 64; r++) {
            int t = (r * u) & 63;
            float ar = z1r[r * HALF + v], ai = z1i[r * HALF + v];
            sr += ar * tc[t] + ai * ts[t];
            si += ai * tc[t] - ar * ts[t];
        }
        size_t o = (size_t)i * CH * BATCH + (size_t)ch * BATCH + b;
        outR[o] = sr; outI[o] = si;
    }
}

// ---------------------------------------------------------------------------
// 4) irfft2 (+ fused pointwise): mode 0: out = sqrt2*relu(sqrt2*z + bias[ch])
//                                mode 1: out = sqrt(1/2)*z + bias[ch]
// ---------------------------------------------------------------------------
__global__ __launch_bounds__(256)
void irfft2_kernel(const float* __restrict__ inR, const float* __restrict__ inI,
                   int CH, int mode, const float* __restrict__ bias,
                   float* __restrict__ dst)
{
    __shared__ float sr[2112], si[2112];
    __shared__ float z1r[2112], z1i[2112];
    __shared__ float tc[64], ts[64];
    int b = blockIdx.x / CH, ch = blockIdx.x % CH;
    int tid = threadIdx.x;
    if (tid < 64) {
        float th = 2.f * PI_F * (float)tid / 64.f;
        tc[tid] = cosf(th); ts[tid] = sinf(th);
    }
    for (int i = tid; i < 2112; i += 256) {
        size_t o = (size_t)i * CH * BATCH + (size_t)ch * BATCH + b;
        sr[i] = inR[o]; si[i] = inI[o];
    }
    __syncthreads();
    // inverse over u: Z1[r][v] = sum_u S[u][v] * e^{+2pi i r u / 64}
    for (int i = tid; i < 2112; i += 256) {
        int r = i / HALF, v = i % HALF;
        float ar = 0.f, ai = 0.f;
        for (int u = 0; u < 64; u++) {
            int t = (r * u) & 63;
            float xr = sr[u * HALF + v], xi = si[u * HALF + v];
            ar += xr * tc[t] - xi * ts[t];
            ai += xr * ts[t] + xi * tc[t];
        }
        z1r[i] = ar; z1i[i] = ai;
    }
    __syncthreads();
    float bv = bias ? bias[ch] : 0.f;
    for (int i = tid; i < 4096; i += 256) {
        int r = i / 64, c = i & 63;
        float acc = z1r[r * HALF + 0];
        acc += (c & 1) ? -z1r[r * HALF + 32] : z1r[r * HALF + 32];
        for (int v = 1; v < 32; v++) {
            int t = (c * v) & 63;
            acc += 2.f * (z1r[r * HALF + v] * tc[t] - z1i[r * HALF + v] * ts[t]);
        }
        acc *= (1.f / 4096.f);
        float o;
        if (mode == 0) o = 1.41421356f * fmaxf(1.41421356f * acc + bv, 0.f);
        else           o = 0.70710678f * acc + bv;
        dst[((size_t)(b * CH + ch)) * 4096 + i] = o;
    }
}

// ---------------------------------------------------------------------------
// 5) Batched per-bin complex GEMM via V_WMMA_F32_16X16X4_F32.
//    C[bin] (MxN) = op(A[bin]) (MxK) @ B[bin] (KxN), N = 64 (batch).
//    CONJT (compile-time): A_logical[m][k] = conj(A_store[k][m]).
//    One wave per 16x16 output tile; split re/im; 4 real WMMAs per K-step,
//    K-loop unrolled x2 (8 WMMAs / iteration) for load/WMMA overlap.
// ---------------------------------------------------------------------------
template <int CONJT>
__global__ __launch_bounds__(256)
void cgemm_wmma_kernel(const float* __restrict__ Are, const float* __restrict__ Aim,
                       int lda, int aBinStride,
                       const float* __restrict__ Bre, const float* __restrict__ Bim,
                       int bBinStride,
                       float* __restrict__ Cre, float* __restrict__ Cim,
                       int cBinStride, int M, int K)
{
    const int N = 64;
    int lane = threadIdx.x & 31;
    int wave = blockIdx.x * 8 + (threadIdx.x >> 5);
    int tilesM = M >> 4;
    int tilesPerBin = tilesM * 4;     // N/16 == 4
    if (wave >= BINS * tilesPerBin) return;
    int bin = wave / tilesPerBin;
    int t   = wave % tilesPerBin;
    int tm  = (t >> 2) << 4;
    int tn  = (t & 3) << 4;

    const float* ar = Are + (size_t)bin * aBinStride;
    const float* ai = Aim + (size_t)bin * aBinStride;
    const float* br = Bre + (size_t)bin * bBinStride;
    const float* bi = Bim + (size_t)bin * bBinStride;
    float* cr = Cre + (size_t)bin * cBinStride;
    float* ci = Cim + (size_t)bin * cBinStride;

    int am   = tm + (lane & 15);
    int kOff = (lane >> 4) << 1;       // 0 (lanes 0-15) or 2 (lanes 16-31)
    int bn   = tn + (lane & 15);

    v8f accRR = {}; v8f accII = {}; v8f accRI = {}; v8f accIR = {};

    #pragma unroll 2
    for (int k0 = 0; k0 < K; k0 += 4) {
        v2f aR, aI, bR, bI;
        if (CONJT == 0) {
            // contiguous k,k+1 -> single 8B vector load each
            aR = *(const v2f*)(ar + (size_t)am * lda + (k0 + kOff));
            aI = *(const v2f*)(ai + (size_t)am * lda + (k0 + kOff));
        } else {
            const float* pr = ar + (size_t)(k0 + kOff) * lda + am;
            const float* pi = ai + (size_t)(k0 + kOff) * lda + am;
            aR[0] = pr[0];   aR[1] = pr[lda];
            aI[0] = -pi[0];  aI[1] = -pi[lda];
        }
        {
            const float* pr = br + (size_t)(k0 + kOff) * N + bn;
            const float* pi = bi + (size_t)(k0 + kOff) * N + bn;
            bR[0] = pr[0]; bR[1] = pr[N];
            bI[0] = pi[0]; bI[1] = pi[N];
        }
        accRR = __builtin_amdgcn_wmma_f32_16x16x4_f32(false, aR, false, bR, (short)0, accRR, false, false);
        accII = __builtin_amdgcn_wmma_f32_16x16x4_f32(false, aI, false, bI, (short)0, accII, false, false);
        accRI = __builtin_amdgcn_wmma_f32_16x16x4_f32(false, aR, false, bI, (short)0, accRI, false, false);
        accIR = __builtin_amdgcn_wmma_f32_16x16x4_f32(false, aI, false, bR, (short)0, accIR, false, false);
    }

    int col   = tn + (lane & 15);
    int rbase = tm + ((lane >> 4) << 3);   // lanes 16-31 hold rows M+8..M+15
    #pragma unroll
    for (int v = 0; v < 8; v++) {
        int row = rbase + v;
        cr[(size_t)row * N + col] = accRR[v] - accII[v];
        ci[(size_t)row * N + col] = accRI[v] + accIR[v];
    }
}

// ---------------------------------------------------------------------------
// 6) Elementwise combine on (bins, rows, 64) slices:
//    out[bin, oOff + e] = a[bin, aOff + e] + sign * b[bin, bOff + e]
// ---------------------------------------------------------------------------
__global__ __launch_bounds__(256)
void ew_combine_kernel(float* __restrict__ oR, float* __restrict__ oI, int osb, int oOff,
                       const float* __restrict__ aR, const float* __restrict__ aI, int asb, int aOff,
                       const float* __restrict__ bR, const float* __restrict__ bI, int bsb, int bOff,
                       float sign, int rows)
{
    int per = rows * 64;
    long long idx = (long long)blockIdx.x * 256 + threadIdx.x;
    if (idx >= (long long)BINS * per) return;
    int bin = (int)(idx / per);
    int rem = (int)(idx % per);
    float vr = aR[(size_t)bin * asb + aOff + rem];
    float vi = aI[(size_t)bin * asb + aOff + rem];
    if (bR) {
        vr += sign * bR[(size_t)bin * bsb + bOff + rem];
        vi += sign * bI[(size_t)bin * bsb + bOff + rem];
    }
    oR[(size_t)bin * osb + oOff + rem] = vr;
    oI[(size_t)bin * osb + oOff + rem] = vi;
}

// ---------------------------------------------------------------------------
// host: kernel_launch
// ---------------------------------------------------------------------------
extern "C" void kernel_launch(void* const* d_in, const int* in_sizes, int n_in,
                              void* d_out, int out_size, void* d_ws, size_t ws_size,
                              hipStream_t stream)
{
    (void)in_sizes; (void)n_in; (void)out_size; (void)ws_size;

    const float* x   = (const float*)d_in[0];
    const float* Fq  = (const float*)d_in[1];
    const float* fq  = (const float*)d_in[2];
    const float* by  = (const float*)d_in[3];
    const float* Fr0 = (const float*)d_in[4];
    const float* fr0 = (const float*)d_in[5];
    const float* b0  = (const float*)d_in[6];
    const float* Fr1 = (const float*)d_in[7];
    const float* fr1 = (const float*)d_in[8];
    const float* b1  = (const float*)d_in[9];
    float* out = (float*)d_out;

    const size_t N128 = (size_t)BINS * 128 * 64;   // 17,301,504 floats
    const size_t N64  = (size_t)BINS * 64 * 64;
    const size_t N32  = (size_t)BINS * 32 * 64;

    size_t off = 0;
    auto alloc = [&](size_t nfloats) -> float* {
        float* p = (float*)((char*)d_ws + off);
        off += ((nfloats * sizeof(float) + 255) / 256) * 256;
        return p;
    };
    float* norm = alloc(64);                  // [0]=Q, [1]=R0, [2]=R1
    float* QR  = alloc(N128); float* QI  = alloc(N128);
    float* R0R = alloc(N64);  float* R0I = alloc(N64);
    float* R1R = alloc(N128); float* R1I = alloc(N128);
    float* xfR = alloc(N32);  float* xfI = alloc(N32);   // xfft, later yfft
    float* xhR = alloc(N128); float* xhI = alloc(N128);
    float* catR= alloc(N128); float* catI= alloc(N128);  // also Ff scratch, also yh
    float* zfR = alloc(N64);  float* zfI = alloc(N64);   // zfft / zfft2
    float* ghR = alloc(N128); float* ghI = alloc(N128);
    float* hkR = alloc(N64);  float* hkI = alloc(N64);
    float* zre = alloc((size_t)BATCH * 64 * 4096);
    float* FfR = catR; float* FfI = catI;                // alias (dead before use)
    float* yhR = catR; float* yhI = catI;                // alias (cat dead by then)

    hipMemsetAsync(norm, 0, 64 * sizeof(float), stream);
    hipFuncSetAttribute((const void*)cayley_kernel,
                        hipFuncAttributeMaxDynamicSharedMemorySize, CAYLEY_LDS_BYTES);

    auto cgemm = [&](const float* Ar_, const float* Ai_, int lda, int asb, int conjT,
                     const float* Br_, const float* Bi_, int bsb,
                     float* Cr_, float* Ci_, int csb, int M, int K) {
        int waves  = BINS * (M / 16) * 4;
        int blocks = (waves + 7) / 8;
        if (conjT)
            cgemm_wmma_kernel<1><<<blocks, 256, 0, stream>>>(Ar_, Ai_, lda, asb,
                                                             Br_, Bi_, bsb, Cr_, Ci_, csb, M, K);
        else
            cgemm_wmma_kernel<0><<<blocks, 256, 0, stream>>>(Ar_, Ai_, lda, asb,
                                                             Br_, Bi_, bsb, Cr_, Ci_, csb, M, K);
    };
    auto ew = [&](float* oR_, float* oI_, int osb, int oOff,
                  const float* aR_, const float* aI_, int asb, int aOff,
                  const float* bR_, const float* bI_, int bsb, int bOff,
                  float sign, int rows) {
        long long cnt = (long long)BINS * rows * 64;
        int blocks = (int)((cnt + 255) / 256);
        ew_combine_kernel<<<blocks, 256, 0, stream>>>(oR_, oI_, osb, oOff,
                                                      aR_, aI_, asb, aOff,
                                                      bR_, bI_, bsb, bOff, sign, rows);
    };

    // ---- weight spectra + Cayley (sequential, Ff scratch reused) ----
    weight_dft_kernel<<<(int)((BINS*128*64 + 255)/256), 256, 0, stream>>>(Fq, 128, 64, FfR, FfI, norm + 0);
    cayley_kernel<<<BINS, 256, CAYLEY_LDS_BYTES, stream>>>(FfR, FfI, fq, norm + 0, 128, 0, 0, QR, QI);

    weight_dft_kernel<<<(int)((BINS*64*64 + 255)/256), 256, 0, stream>>>(Fr0, 64, 64, FfR, FfI, norm + 1);
    cayley_kernel<<<BINS, 256, CAYLEY_LDS_BYTES, stream>>>(FfR, FfI, fr0, norm + 1, 64, 0, 0, R0R, R0I);

    weight_dft_kernel<<<(int)((BINS*64*128 + 255)/256), 256, 0, stream>>>(Fr1, 64, 128, FfR, FfI, norm + 2);
    cayley_kernel<<<BINS, 256, CAYLEY_LDS_BYTES, stream>>>(FfR, FfI, fr1, norm + 2, 128, 1, 1, R1R, R1I);

    // ---- x spectrum (sqrt(GAMMA)=1) ----
    rfft2_fwd_kernel<<<BATCH * 32, 256, 0, stream>>>(x, 32, 1.0f, xfR, xfI);

    // ---- xhfft = Qfft[:, :, :32] @ xfft : M=128, K=32 ----
    cgemm(QR, QI, 64, 8192, 0, xfR, xfI, 2048, xhR, xhI, 8192, 128, 32);

    // ---- layer 0: zfft = R0 @ xh[0:64] ----
    cgemm(R0R, R0I, 64, 4096, 0, xhR, xhI, 8192, zfR, zfI, 4096, 64, 64);
    irfft2_kernel<<<BATCH * 64, 256, 0, stream>>>(zfR, zfI, 64, 0, b0, zre);
    rfft2_fwd_kernel<<<BATCH * 64, 256, 0, stream>>>(zre, 64, 1.0f, zfR, zfI);
    // gh0 = R0^H @ zfft2 : M=64, K=64  (gh stored (bins,64,64) here)
    cgemm(R0R, R0I, 64, 4096, 1, zfR, zfI, 4096, ghR, ghI, 4096, 64, 64);
    // hk0 = gh0 - xh[0:64]
    ew(hkR, hkI, 4096, 0, ghR, ghI, 4096, 0, xhR, xhI, 8192, 0, -1.f, 64);

    // ---- layer 1: cat = [xh[64:128] ; hk0] ----
    ew(catR, catI, 8192, 0,    xhR, xhI, 8192, 4096, nullptr, nullptr, 0, 0, 0.f, 64);
    ew(catR, catI, 8192, 4096, hkR, hkI, 4096, 0,    nullptr, nullptr, 0, 0, 0.f, 64);
    // zfft = R1 @ cat : M=64, K=128 (R1 stored (bins,64,128), lda=128)
    cgemm(R1R, R1I, 128, 8192, 0, catR, catI, 8192, zfR, zfI, 4096, 64, 128);
    irfft2_kernel<<<BATCH * 64, 256, 0, stream>>>(zfR, zfI, 64, 0, b1, zre);
    rfft2_fwd_kernel<<<BATCH * 64, 256, 0, stream>>>(zre, 64, 1.0f, zfR, zfI);
    // gh1 = R1^H @ zfft2 : M=128, K=64 (gh stored (bins,128,64))
    cgemm(R1R, R1I, 128, 8192, 1, zfR, zfI, 4096, ghR, ghI, 8192, 128, 64);

    // ---- yhfft: rows 0:64 = hk0 - gh1[64:128]; rows 64:128 = gh1[0:64] - xh[64:128]
    ew(yhR, yhI, 8192, 0,    hkR, hkI, 4096, 0, ghR, ghI, 8192, 4096, -1.f, 64);
    ew(yhR, yhI, 8192, 4096, ghR, ghI, 8192, 0, xhR, xhI, 8192, 4096, -1.f, 64);
    // t = xhfft + yhfft (in place into yh)
    ew(yhR, yhI, 8192, 0, yhR, yhI, 8192, 0, xhR, xhI, 8192, 0, 1.f, 128);

    // ---- yfft = conj(Qfft[:, :, 32:])^T @ t : M=32, K=128 (reuse xf buffers) ----
    cgemm(QR + 32, QI + 32, 64, 8192, 1, yhR, yhI, 8192, xfR, xfI, 2048, 32, 128);

    // ---- y = sqrt(1/2) * irfft2(yfft) + by ----
    irfft2_kernel<<<BATCH * 32, 256, 0, stream>>>(xfR, xfI, 32, 1, by, out);
}